// GQA_SWA_Head_33225867002141
// MI455X (gfx1250) — compile-verified
//
#include <hip/hip_runtime.h>
#include <hip/hip_bf16.h>
#include <stdint.h>

typedef float v2f __attribute__((ext_vector_type(2)));
typedef float v8f __attribute__((ext_vector_type(8)));
typedef unsigned int u32x4 __attribute__((ext_vector_type(4)));
typedef int i32x4 __attribute__((ext_vector_type(4)));
typedef int i32x8 __attribute__((ext_vector_type(8)));

// 16-byte generic vector type matching the async-LDS builtin's pointee type
typedef int vs4i __attribute__((vector_size(16)));
typedef __attribute__((address_space(1))) vs4i g_vs4i;   // global
typedef __attribute__((address_space(3))) vs4i l_vs4i;   // LDS

#define T_SEQ 2048
#define NE    1024
#define DH    128
#define NB    4

// ---- feature probes (graceful fallback keeps compile green) ----
#if defined(__has_builtin)
#if __has_builtin(__builtin_amdgcn_global_load_async_to_lds_b128) && \
    __has_builtin(__builtin_amdgcn_s_wait_asynccnt)
#define HAVE_ASYNC_LDS 1
#endif
#if __has_builtin(__builtin_amdgcn_tensor_load_to_lds) && \
    __has_builtin(__builtin_amdgcn_s_wait_tensorcnt)
#define HAVE_TDM 1
#endif
#endif

typedef __attribute__((address_space(3))) void las_void;

__device__ __forceinline__ void async_copy16(const float* gsrc, float* ldst) {
#if defined(HAVE_ASYNC_LDS)
  __builtin_amdgcn_global_load_async_to_lds_b128(
      (g_vs4i*)gsrc, (l_vs4i*)ldst, 0, 0);
#else
  *(float4*)ldst = *(const float4*)gsrc;
#endif
}

__device__ __forceinline__ void async_wait_all() {
#if defined(HAVE_ASYNC_LDS)
  __builtin_amdgcn_s_wait_asynccnt(0);
#endif
}

// wait until only the newest tile's 8 async ops (per wave) may be outstanding
__device__ __forceinline__ void async_wait_keep8() {
#if defined(HAVE_ASYNC_LDS)
  __builtin_amdgcn_s_wait_asynccnt(8);
#endif
}

__device__ __forceinline__ v8f wmma_f32(v2f a, v2f b, v8f c) {
  // D = A(16x4,f32) x B(4x16,f32) + C(16x16,f32)
  return __builtin_amdgcn_wmma_f32_16x16x4_f32(false, a, false, b, (short)0, c, false, false);
}

// ---------------- Stage 0: fold 8 query heads: Wqs = sum_h Wq[h] ----------------
__global__ __launch_bounds__(256) void wq_sum_kernel(const float* __restrict__ Wq,
                                                     float* __restrict__ Wqs) {
  int i = blockIdx.x * 256 + threadIdx.x;   // < 1024*128
  float s = 0.f;
#pragma unroll
  for (int h = 0; h < 8; ++h) s += Wq[h * (NE * DH) + i];
  Wqs[i] = s;
}

// ---------------- Stage 1: Out[M][128] = X[M][1024] * W[1024][128] ----------------
// Block: 128 rows x 128 cols, 8 waves; wave w -> rows [16w,16w+16), 8 n-tiles.
// Double-buffered LDS tiles; async global->LDS staging overlapped with WMMA.
#define XS_STRIDE 36
#define WS_STRIDE 136
#define XBUF_SZ   (128 * XS_STRIDE)
#define WBUF_SZ   (32 * WS_STRIDE)

__device__ __forceinline__ void stage_tile(const float* __restrict__ X,
                                           const float* __restrict__ W,
                                           int m0, int kk, int tid,
                                           float* xbuf, float* wbuf) {
  // X tile [128][32] -> xbuf (stride 36), 4 async b128 per thread
#pragma unroll
  for (int q = 0; q < 4; ++q) {
    int idx = tid + 256 * q;              // 0..1023 float4s
    int row = idx >> 3;
    int c4  = (idx & 7) << 2;
    async_copy16(X + (m0 + row) * NE + kk + c4, xbuf + row * XS_STRIDE + c4);
  }
  // W tile [32][128] -> wbuf (stride 136), 4 async b128 per thread
#pragma unroll
  for (int q = 0; q < 4; ++q) {
    int idx = tid + 256 * q;
    int row = idx >> 5;
    int c4  = (idx & 31) << 2;
    async_copy16(W + (kk + row) * DH + c4, wbuf + row * WS_STRIDE + c4);
  }
}

__global__ __launch_bounds__(256) void qkv_gemm_kernel(const float* __restrict__ X,
                                                       const float* __restrict__ W,
                                                       float* __restrict__ Out) {
  extern __shared__ float qsmem[];        // 2*XBUF_SZ + 2*WBUF_SZ floats

  const int tid  = threadIdx.x;
  const int wave = tid >> 5;
  const int lane = tid & 31;
  const int half = lane >> 4;
  const int lid  = lane & 15;
  const int m0   = blockIdx.x * 128;

  v8f acc[8];
#pragma unroll
  for (int nt = 0; nt < 8; ++nt) acc[nt] = (v8f){0,0,0,0,0,0,0,0};

  // prologue: stage first tile into buffer 0
  stage_tile(X, W, m0, 0, tid, qsmem, qsmem + 2 * XBUF_SZ);

  for (int kk = 0; kk < NE; kk += 32) {
    const int cur = (kk >> 5) & 1;
    const int nxt = cur ^ 1;
    if (kk + 32 < NE) {
      // prefetch next tile, then wait only for the previous tile's 8 ops
      stage_tile(X, W, m0, kk + 32, tid,
                 qsmem + nxt * XBUF_SZ,
                 qsmem + 2 * XBUF_SZ + nxt * WBUF_SZ);
      async_wait_keep8();
    } else {
      async_wait_all();
    }
    __syncthreads();

    const float* xb = qsmem + cur * XBUF_SZ;
    const float* wb = qsmem + 2 * XBUF_SZ + cur * WBUF_SZ;
#pragma unroll
    for (int k4 = 0; k4 < 32; k4 += 4) {
      v2f a;
      a.x = xb[(wave * 16 + lid) * XS_STRIDE + k4 + 2 * half];
      a.y = xb[(wave * 16 + lid) * XS_STRIDE + k4 + 2 * half + 1];
#pragma unroll
      for (int nt = 0; nt < 8; ++nt) {
        v2f b;
        b.x = wb[(k4 + 2 * half) * WS_STRIDE + nt * 16 + lid];
        b.y = wb[(k4 + 2 * half + 1) * WS_STRIDE + nt * 16 + lid];
        acc[nt] = wmma_f32(a, b, acc[nt]);
      }
    }
    __syncthreads();   // all reads of buf[cur] done before it is restaged
  }
#pragma unroll
  for (int nt = 0; nt < 8; ++nt)
#pragma unroll
    for (int r = 0; r < 8; ++r)
      Out[(m0 + wave * 16 + r + 8 * half) * DH + nt * 16 + lid] = acc[nt][r];
}

// ---------------- Stage 2: fused attention per (b, 16-row t-tile) ----------------
// S = Q_tile[16x128] x Kflat[128x2048] (the reshape "bug" == flat reinterpretation),
// masked softmax over s, out_tile = P[16x2048] x V[2048x128].
__global__ __launch_bounds__(256) void attn_kernel(const float* __restrict__ Q,
                                                   const float* __restrict__ K,
                                                   const float* __restrict__ V,
                                                   float* __restrict__ out,
                                                   const int* __restrict__ win_p) {
  extern __shared__ float smem[];
  float* S      = smem;               // [16][2049]
  float* Qs     = S + 16 * 2049;      // [16][132]  (row stride 132 floats)
  float* red    = Qs + 16 * 132;      // [16][16]
  float* rowsum = red + 256;          // [16]

  const int tid  = threadIdx.x;
  const int wave = tid >> 5;
  const int lane = tid & 31;
  const int half = lane >> 4;
  const int lid  = lane & 15;
  const int b    = blockIdx.x >> 7;
  const int t0   = (blockIdx.x & 127) << 4;
  const int WIN  = *win_p;
  const float scale = 0.088388347648318447f;   // 128^-0.5

  const float* Qb = Q + b * (T_SEQ * DH);
  const float* Kb = K + b * (T_SEQ * DH);      // viewed flat as [128][2048]
  const float* Vb = V + b * (T_SEQ * DH);

  // ---- stage Q tile [16][128] into Qs (stride 132) ----
#if defined(HAVE_TDM)
  if (wave == 0) {
    // D# group 0: count=1, lds_addr, 57-bit global addr, type=2
    uint64_t gaddr = (uint64_t)(uintptr_t)(Qb + t0 * DH);
    uint32_t laddr = (uint32_t)(uintptr_t)(las_void*)Qs;
    u32x4 g0;
    g0[0] = 1u;                                  // count=1, no gather
    g0[1] = laddr;                               // lds_addr (bytes)
    g0[2] = (uint32_t)(gaddr & 0xFFFFFFFFu);     // global_addr[31:0]
    g0[3] = (uint32_t)((gaddr >> 32) & 0x1FFFFFFu) | 0x80000000u; // addr[56:32] | type=2
    // D# group 1: data_size=4B, pad_enable, pad every 128 DW by 4 DW,
    // tensor 128x16 (dim0 stride 128), tile 128x16
    i32x8 g1;
    g1[0] = (2 << 16) | (1 << 20) | (6 << 22) | (3 << 25);
    g1[1] = (int)((128u & 0xFFFFu) << 16);       // tensor_dim0 lo16 << 16
    g1[2] = (int)(16u << 16);                    // tensor_dim0 hi16=0 | tensor_dim1 lo16
    g1[3] = (int)(128u << 16);                   // tensor_dim1 hi16=0 | tile_dim0=128
    g1[4] = 16;                                  // tile_dim1=16, tile_dim2=0
    g1[5] = 128;                                 // tensor_dim0_stride lo32
    g1[6] = 0;                                   // stride0 hi | stride1 lo
    g1[7] = 0;
    i32x4 z4 = {0, 0, 0, 0};
#if __has_include(<hip/amd_detail/amd_gfx1250_TDM.h>)
    i32x8 z8 = {0, 0, 0, 0, 0, 0, 0, 0};
    __builtin_amdgcn_tensor_load_to_lds(g0, g1, z4, z4, z8, 0);
#else
    __builtin_amdgcn_tensor_load_to_lds(g0, g1, z4, z4, 0);
#endif
    __builtin_amdgcn_s_wait_tensorcnt(0);
  }
#else
#pragma unroll
  for (int q = 0; q < 2; ++q) {
    int idx = tid + 256 * q;          // 0..511 float4s
    int row = idx >> 5;
    int c4  = (idx & 31) << 2;
    float4 v = *(const float4*)(Qb + (t0 + row) * DH + c4);
    *(float4*)(&Qs[row * 132 + c4]) = v;
  }
#endif
  __syncthreads();

  // preload all 32 A-fragments for the 128-deep contraction (2 floats each)
  float afr[64];
#pragma unroll
  for (int k = 0; k < 32; ++k) {
    afr[2 * k]     = Qs[lid * 132 + 4 * k + 2 * half];
    afr[2 * k + 1] = Qs[lid * 132 + 4 * k + 2 * half + 1];
  }

  // scores: wave handles s-tiles (wave + 8*i), i = 0..15
  for (int i = 0; i < 16; ++i) {
    int s0 = (wave + 8 * i) * 16;
    v8f acc = {0, 0, 0, 0, 0, 0, 0, 0};
    __builtin_prefetch(Kb + 2 * half * T_SEQ + ((s0 + 128) & (T_SEQ - 1)), 0, 1);
#pragma unroll
    for (int k = 0; k < 32; ++k) {
      v2f a; a.x = afr[2 * k]; a.y = afr[2 * k + 1];
      const float* bp = Kb + (4 * k + 2 * half) * T_SEQ + s0 + lid;
      v2f bf; bf.x = bp[0]; bf.y = bp[T_SEQ];
      acc = wmma_f32(a, bf, acc);
    }
#pragma unroll
    for (int r = 0; r < 8; ++r)
      S[(r + 8 * half) * 2049 + s0 + lid] = acc[r];
  }
  __syncthreads();

  // masked softmax (keep P unnormalized; fold 1/rowsum into the output store)
  const int row = tid & 15;
  const int j   = tid >> 4;
  const int t   = t0 + row;
  float m = -__builtin_inff();
  for (int s = j; s < T_SEQ; s += 16) {
    if (s <= t - WIN) continue;                 // past-window mask (future unmasked)
    m = fmaxf(m, S[row * 2049 + s] * scale);
  }
  red[row * 16 + j] = m;
  __syncthreads();
  float rmax = -__builtin_inff();
#pragma unroll
  for (int q = 0; q < 16; ++q) rmax = fmaxf(rmax, red[row * 16 + q]);
  __syncthreads();
  float lsum = 0.f;
  for (int s = j; s < T_SEQ; s += 16) {
    float p = (s <= t - WIN) ? 0.f : __expf(S[row * 2049 + s] * scale - rmax);
    S[row * 2049 + s] = p;
    lsum += p;
  }
  red[row * 16 + j] = lsum;
  __syncthreads();
  float rsum = 0.f;
#pragma unroll
  for (int q = 0; q < 16; ++q) rsum += red[row * 16 + q];
  if (tid < 16) rowsum[tid] = rsum;
  __syncthreads();

  // out tile: wave -> n-tile, contraction over s = 2048
  {
    const int nt = wave;
    v8f acc = {0, 0, 0, 0, 0, 0, 0, 0};
#pragma unroll 4
    for (int k = 0; k < 512; ++k) {
      int kb = 4 * k + 2 * half;
      v2f a; a.x = S[lid * 2049 + kb]; a.y = S[lid * 2049 + kb + 1];
      const float* vp = Vb + kb * DH + nt * 16 + lid;
      v2f bf; bf.x = vp[0]; bf.y = vp[DH];
      acc = wmma_f32(a, bf, acc);
    }
#pragma unroll
    for (int r = 0; r < 8; ++r) {
      int rr = r + 8 * half;
      out[(b * T_SEQ + t0 + rr) * DH + nt * 16 + lid] = acc[r] / rowsum[rr];
    }
  }
}

extern "C" void kernel_launch(void* const* d_in, const int* in_sizes, int n_in,
                              void* d_out, int out_size, void* d_ws, size_t ws_size,
                              hipStream_t stream) {
  const float* x  = (const float*)d_in[0];
  const float* Wk = (const float*)d_in[1];
  const float* Wq = (const float*)d_in[2];
  const float* Wv = (const float*)d_in[3];
  const int*  win = (const int*)d_in[4];

  float* Kf  = (float*)d_ws;                 // [B,T,128]  4 MB
  float* Qf  = Kf + NB * T_SEQ * DH;         // [B,T,128]  4 MB
  float* Vf  = Qf + NB * T_SEQ * DH;         // [B,T,128]  4 MB
  float* Wqs = Vf + NB * T_SEQ * DH;         // [1024,128] 512 KB

  wq_sum_kernel<<<(NE * DH) / 256, 256, 0, stream>>>(Wq, Wqs);

  size_t qsmem = (size_t)(2 * XBUF_SZ + 2 * WBUF_SZ) * sizeof(float);  // ~70 KB
  qkv_gemm_kernel<<<(NB * T_SEQ) / 128, 256, qsmem, stream>>>(x, Wk,  Kf);
  qkv_gemm_kernel<<<(NB * T_SEQ) / 128, 256, qsmem, stream>>>(x, Wqs, Qf);
  qkv_gemm_kernel<<<(NB * T_SEQ) / 128, 256, qsmem, stream>>>(x, Wv,  Vf);

  size_t smem = (size_t)(16 * 2049 + 16 * 132 + 256 + 16) * sizeof(float);
  attn_kernel<<<NB * (T_SEQ / 16), 256, smem, stream>>>(Qf, Kf, Vf, (float*)d_out, win);
}